// MyModel_61933428410727
// MI455X (gfx1250) — compile-verified
//
#include <hip/hip_runtime.h>
#include <hip/hip_bf16.h>
#include <stdint.h>

#ifndef __has_builtin
#define __has_builtin(x) 0
#endif

// Problem dims (fixed by reference setup_inputs)
#define CTC_T 512
#define CTC_N 32
#define CTC_C 4000
#define CTC_S 64
#define CTC_L (2 * CTC_S + 1) /* 129 */

#define NEG_INF (-1e30f)
#define GUARD (-5e29f) /* NEG_INF/2 */

// ---------------------------------------------------------------------------
// gfx1250 async global->LDS DMA helpers (ASYNCcnt-tracked).
// Probe round 1 revealed the builtin signatures:
//   b128: (v4i32 AS1*, <lds> AS3*, imm int offset, imm int cpol)
//   b32 : (i32   AS1*, <lds> AS3*, imm int offset, imm int cpol)
// (clang prints AS1 as `__device__` in HIP diagnostics)
// ---------------------------------------------------------------------------
#if defined(__AMDGCN__) && __has_builtin(__builtin_amdgcn_global_load_async_to_lds_b128)
#define HAVE_ASYNC_LDS 1
#else
#define HAVE_ASYNC_LDS 0
#endif

typedef int v4i __attribute__((ext_vector_type(4)));
#define AS1 __attribute__((address_space(1)))
#define AS3 __attribute__((address_space(3)))

__device__ __forceinline__ void async_g2l_b128(const float* g, float* l) {
#if HAVE_ASYNC_LDS
  __builtin_amdgcn_global_load_async_to_lds_b128(
      (AS1 v4i*)(AS1 float*)const_cast<float*>(g),
      (AS3 v4i*)(AS3 float*)l, 0, 0);
#else
  *(float4*)l = *(const float4*)g;
#endif
}

__device__ __forceinline__ void async_g2l_b32(const float* g, float* l) {
#if HAVE_ASYNC_LDS && __has_builtin(__builtin_amdgcn_global_load_async_to_lds_b32)
  __builtin_amdgcn_global_load_async_to_lds_b32(
      (AS1 int*)(AS1 float*)const_cast<float*>(g),
      (AS3 int*)(AS3 float*)l, 0, 0);
#else
  *l = *g;
#endif
}

__device__ __forceinline__ void wait_async_zero() {
#if HAVE_ASYNC_LDS
#if __has_builtin(__builtin_amdgcn_s_wait_asynccnt)
  __builtin_amdgcn_s_wait_asynccnt(0);
#else
  asm volatile("s_wait_asynccnt 0" ::: "memory");
#endif
#endif
  asm volatile("" ::: "memory");
}

// ---------------------------------------------------------------------------
// Kernel A: per (t,n) row: stage 4000-float row into LDS via async DMA,
// compute logsumexp (max pass + sum-exp pass, both from LDS => global traffic
// stays 1x), then gather the 129 extended-label log-probs from the LDS copy.
// One wave per row; 16384 rows total. Memory-bound: 262 MB @ 23.3 TB/s.
// ---------------------------------------------------------------------------
__global__ __launch_bounds__(32) void ctc_lse_gather(const float* __restrict__ x,
                                                     const int* __restrict__ labels,
                                                     float* __restrict__ lp_ext) {
  __shared__ float buf[4096]; // 16 KB; row is 4000 floats
  const int row = blockIdx.x;       // row = t*N + n
  const int n = row & (CTC_N - 1);  // N = 32
  const int lane = threadIdx.x;
  const float* xr = x + (size_t)row * CTC_C;

  // Stage: 31 issues of b128 (512 B / wave) cover 3968 floats; b32 covers tail 32.
#pragma unroll
  for (int k = 0; k < 31; ++k) {
    const int fi = k * 128 + lane * 4;
    async_g2l_b128(xr + fi, &buf[fi]);
  }
  async_g2l_b32(xr + 3968 + lane, &buf[3968 + lane]);
  wait_async_zero();

  // Pass 1: row max (lane-strided, conflict-free: bank = (lane + 32j) % 64)
  float m = NEG_INF;
#pragma unroll 5
  for (int j = 0; j < 125; ++j) m = fmaxf(m, buf[lane + 32 * j]);
#pragma unroll
  for (int off = 16; off > 0; off >>= 1) m = fmaxf(m, __shfl_xor(m, off, 32));

  // Pass 2: sum of exp(x - max)
  float s = 0.f;
#pragma unroll 5
  for (int j = 0; j < 125; ++j) s += __expf(buf[lane + 32 * j] - m);
#pragma unroll
  for (int off = 16; off > 0; off >>= 1) s += __shfl_xor(s, off, 32);

  const float lse = m + __logf(s);

  // Gather extended labels: even l -> blank(class 0), odd l -> labels[n][(l-1)/2]
  float* outp = lp_ext + (size_t)row * CTC_L;
  for (int l = lane; l < CTC_L; l += 32) {
    const int lab = (l & 1) ? labels[n * CTC_S + (l >> 1)] : 0;
    outp[l] = buf[lab] - lse;
  }
}

// ---------------------------------------------------------------------------
// Kernel B: sequential CTC alpha recurrence. One wave per batch item; the 129
// alpha cells live in 5 registers/lane (l = lane + 32k). Neighbor values come
// from wave32 shuffles -> zero barriers across the 512 dependent steps.
// ---------------------------------------------------------------------------
__global__ __launch_bounds__(32) void ctc_alpha(const float* __restrict__ lp_ext,
                                                const int* __restrict__ labels,
                                                const int* __restrict__ input_lengths,
                                                const int* __restrict__ label_lengths,
                                                float* __restrict__ losses) {
  const int n = blockIdx.x;
  const int lane = threadIdx.x;
  const int Tin = input_lengths[n];
  const int Sl = label_lengths[n];

  bool valid[5], skipok[5];
  float a[5];
#pragma unroll
  for (int k = 0; k < 5; ++k) {
    const int l = lane + 32 * k;
    valid[k] = (l < CTC_L);
    int lab = 0, labm2 = -1;
    if ((l & 1) && valid[k]) {
      lab = labels[n * CTC_S + (l >> 1)];
      if (l >= 3) labm2 = labels[n * CTC_S + ((l - 3) >> 1)];
    }
    skipok[k] = ((l & 1) != 0) && valid[k] && (lab != 0) && (lab != labm2);
    // alpha at t=0: only l=0 and (l=1 if Sl>0) start finite
    const float lp0 = valid[k] ? lp_ext[(size_t)n * CTC_L + l] : NEG_INF;
    float v = NEG_INF;
    if (l == 0) v = lp0;
    else if (l == 1) v = (Sl > 0) ? lp0 : NEG_INF;
    a[k] = v;
  }

  for (int t = 1; t < CTC_T; ++t) {
    const float* lpt = lp_ext + ((size_t)t * CTC_N + n) * CTC_L;
    float lp[5];
#pragma unroll
    for (int k = 0; k < 5; ++k) {
      const int l = lane + 32 * k;
      lp[k] = lpt[valid[k] ? l : 0];
    }
    if (t + 8 < CTC_T)
      __builtin_prefetch(lp_ext + ((size_t)(t + 8) * CTC_N + n) * CTC_L + lane * 4, 0, 1);

    // Snapshot cross-lane neighbors of the OLD alpha before any update.
    float up1[5], up2[5], hi1[5], hi2[5];
#pragma unroll
    for (int k = 0; k < 5; ++k) {
      up1[k] = __shfl_up(a[k], 1, 32);
      up2[k] = __shfl_up(a[k], 2, 32);
      hi1[k] = __shfl(a[k], 31, 32); // value at l = 32k + 31
      hi2[k] = __shfl(a[k], 30, 32); // value at l = 32k + 30
    }
    const bool active = (t < Tin);
#pragma unroll
    for (int k = 0; k < 5; ++k) {
      float am1 = (lane == 0) ? (k > 0 ? hi1[k - 1] : NEG_INF) : up1[k];
      float am2;
      if (lane >= 2) am2 = up2[k];
      else if (k == 0) am2 = NEG_INF;
      else am2 = (lane == 0) ? hi2[k - 1] : hi1[k - 1];
      if (!skipok[k]) am2 = NEG_INF;

      const float a0 = a[k];
      const float mx = fmaxf(a0, fmaxf(am1, am2));
      const float lse =
          mx + __logf(__expf(a0 - mx) + __expf(am1 - mx) + __expf(am2 - mx));
      float nv = (mx > GUARD) ? (lse + lp[k]) : NEG_INF;
      if (!valid[k]) nv = NEG_INF;
      a[k] = active ? nv : a0;
    }
  }

  // loss_n = -LSE(alpha[2*Sl], alpha[2*Sl-1])
  const int idx = 2 * Sl;
  const int idx2 = (idx > 0) ? idx - 1 : 0;
  float c1[5], c2[5];
#pragma unroll
  for (int k = 0; k < 5; ++k) {
    c1[k] = __shfl(a[k], idx & 31, 32);
    c2[k] = __shfl(a[k], idx2 & 31, 32);
  }
  float alast = c1[0], aprev = c2[0];
#pragma unroll
  for (int k = 1; k < 5; ++k) {
    if ((idx >> 5) == k) alast = c1[k];
    if ((idx2 >> 5) == k) aprev = c2[k];
  }
  if (Sl <= 0) aprev = NEG_INF;
  const float mx = fmaxf(alast, aprev);
  float loss = -(mx + __logf(__expf(alast - mx) + __expf(aprev - mx)));
  if (!(loss < -GUARD)) loss = 0.f; // zero_infinity (also maps NaN -> 0)
  if (lane == 0) losses[n] = loss;
}

// ---------------------------------------------------------------------------
// Kernel C: deterministic reduction of the 32 per-item losses (no atomics).
// ---------------------------------------------------------------------------
__global__ __launch_bounds__(32) void ctc_sum(const float* __restrict__ losses,
                                              float* __restrict__ out) {
  float v = losses[threadIdx.x];
#pragma unroll
  for (int off = 16; off > 0; off >>= 1) v += __shfl_xor(v, off, 32);
  if (threadIdx.x == 0) out[0] = v;
}

extern "C" void kernel_launch(void* const* d_in, const int* in_sizes, int n_in,
                              void* d_out, int out_size, void* d_ws, size_t ws_size,
                              hipStream_t stream) {
  (void)in_sizes; (void)n_in; (void)out_size; (void)ws_size;
  const float* x = (const float*)d_in[0];        // (T, N, C) fp32 logits
  const int* labels = (const int*)d_in[1];       // (N, S)
  const int* in_len = (const int*)d_in[2];       // (N,)
  const int* lab_len = (const int*)d_in[3];      // (N,)

  // Workspace: lp_ext (T*N*L floats = 8.45 MB) + losses (N floats)
  float* lp_ext = (float*)d_ws;
  float* losses = lp_ext + (size_t)CTC_T * CTC_N * CTC_L;
  float* out = (float*)d_out;

  ctc_lse_gather<<<CTC_T * CTC_N, 32, 0, stream>>>(x, labels, lp_ext);
  ctc_alpha<<<CTC_N, 32, 0, stream>>>(lp_ext, labels, in_len, lab_len, losses);
  ctc_sum<<<1, 32, 0, stream>>>(losses, out);
}